// CrossPixelRefinement_88957362635039
// MI455X (gfx1250) — compile-verified
//
#include <hip/hip_runtime.h>
#include <math.h>

typedef float v2f __attribute__((ext_vector_type(2)));
typedef float v8f __attribute__((ext_vector_type(8)));

#define HASH_EMPTY (-1)
#define WPB 8  // waves per block (wave32)

__device__ __forceinline__ float gelu_tanh(float x) {
    const float k = 0.7978845608028654f;  // sqrt(2/pi)
    float x3 = x * x * x;
    return 0.5f * x * (1.0f + tanhf(k * (x + 0.044715f * x3)));
}

// ---------------------------------------------------------------------------
// Fold w3 (1x7) * w2 (7x1) * w1 (1x1) into 49 tap matrices T[tap] in R^{8x2},
// stored padded to [50 taps][16 rows][2 cols] for direct WMMA A-fragment use.
// Also folds biases into the interior-constant c3[8] (exact: biases are zero).
// ---------------------------------------------------------------------------
__global__ void cpr_prep(const float* __restrict__ w1, const float* __restrict__ b1,
                         const float* __restrict__ w2, const float* __restrict__ b2,
                         const float* __restrict__ w3, const float* __restrict__ b3,
                         float* __restrict__ Tg /*1600 floats*/,
                         float* __restrict__ c3g /*8 floats*/) {
    int tid = threadIdx.x;
    for (int e = tid; e < 1600; e += blockDim.x) {
        int tap = e >> 5;        // [tap][row][c] with row stride 2
        int rem = e & 31;
        int row = rem >> 1;
        int c   = rem & 1;
        float v = 0.0f;
        if (tap < 49 && row < 8) {
            int kh = tap / 7, kw = tap % 7;   // kh: vertical (w2), kw: horizontal (w3)
            for (int i = 0; i < 8; ++i) {
                float a3 = w3[row * 56 + i * 7 + kw];   // w3[row][i][0][kw]
                float s = 0.0f;
                for (int m = 0; m < 8; ++m)
                    s += w2[i * 56 + m * 7 + kh] * w1[m * 2 + c];  // w2[i][m][kh][0]*w1[m][c]
                v += a3 * s;
            }
        }
        Tg[e] = v;
    }
    if (tid < 8) {
        int o = tid;
        float acc = b3[o];
        for (int i = 0; i < 8; ++i) {
            float sw3 = 0.0f;
            for (int kw = 0; kw < 7; ++kw) sw3 += w3[o * 56 + i * 7 + kw];
            float u = b2[i];
            for (int m = 0; m < 8; ++m) {
                float sw2 = 0.0f;
                for (int kh = 0; kh < 7; ++kh) sw2 += w2[i * 56 + m * 7 + kh];
                u += sw2 * b1[m];
            }
            acc += sw3 * u;
        }
        c3g[o] = acc;
    }
}

__global__ void cpr_clear(int* __restrict__ keys, int cap) {
    int i = blockIdx.x * blockDim.x + threadIdx.x;
    if (i < cap) keys[i] = HASH_EMPTY;
}

// ---------------------------------------------------------------------------
// Insert scatter points into open-addressing hash: key = flat pixel id,
// value = the two grid channel values (fc1 / s1).
// ---------------------------------------------------------------------------
__global__ void cpr_insert(const float* __restrict__ fc0, const float* __restrict__ fc1,
                           const int* __restrict__ bidx,
                           const float* __restrict__ scale0, const float* __restrict__ scale1,
                           int* __restrict__ keys, float2* __restrict__ vals, int N,
                           const int* __restrict__ h0p, const int* __restrict__ w0p,
                           const int* __restrict__ fsp, unsigned hmask) {
    int n = blockIdx.x * blockDim.x + threadIdx.x;
    if (n >= N) return;
    int H = *h0p, W = *w0p;
    float FSf = (float)(*fsp);
    int b = bidx[n];
    float s0x = scale0[2 * b] * FSf, s0y = scale0[2 * b + 1] * FSf;
    int x = (int)rintf(fc0[2 * n] / s0x - 0.5f);
    int y = (int)rintf(fc0[2 * n + 1] / s0y - 0.5f);
    int pid = (b * H + y) * W + x;
    float v0 = fc1[2 * n]     / (scale1[2 * b] * FSf);
    float v1 = fc1[2 * n + 1] / (scale1[2 * b + 1] * FSf);
    unsigned slot = ((unsigned)pid * 2654435761u) & hmask;
    for (;;) {
        int old = atomicCAS(&keys[slot], HASH_EMPTY, pid);
        if (old == HASH_EMPTY || old == pid) break;
        slot = (slot + 1) & hmask;
    }
    float2 v; v.x = v0; v.y = v1;
    vals[slot] = v;
}

// ---------------------------------------------------------------------------
// Main: one wave32 per 16 points. Probe the 49-tap window into dense LDS
// V[50][2][16], then H3[8x16] = T * V via v_wmma_f32_16x16x4_f32 (K=4 packs
// 2 taps), skipping all-empty tap pairs with a scalar branch (EXEC stays ~0).
// Epilogue (lanes 0-15, one point each): gelu, 1x1 conv w4, residual, *s1.
// ---------------------------------------------------------------------------
__global__ __launch_bounds__(256) void cpr_main(
    const float* __restrict__ fc0, const int* __restrict__ bidx,
    const float* __restrict__ scale0, const float* __restrict__ scale1,
    const float* __restrict__ w4, const float* __restrict__ b4,
    const float* __restrict__ Tg, const float* __restrict__ c3g,
    const int* __restrict__ keys, const float2* __restrict__ vals,
    float* __restrict__ out, int N,
    const int* __restrict__ h0p, const int* __restrict__ w0p,
    const int* __restrict__ fsp, unsigned hmask)
{
    __shared__ float Tlds[1600];          // [50][16][2], A-matrix taps (rows 8..15 zero)
    __shared__ float Vw[WPB][1600];       // [50][2][16] window values per wave
    __shared__ int   occ[WPB][64];        // per-tap occupancy flags
    __shared__ int   pX[WPB][16], pY[WPB][16], pB[WPB][16];

    const int H = *h0p, W = *w0p;
    const float FSf = (float)(*fsp);

    int tid  = threadIdx.x;
    int wave = tid >> 5, lane = tid & 31;
    int group = blockIdx.x * WPB + wave;

    for (int e = tid; e < 1600; e += 256) Tlds[e] = Tg[e];
    for (int e = lane; e < 1600; e += 32) Vw[wave][e] = 0.0f;
    for (int e = lane; e < 64; e += 32) occ[wave][e] = 0;

    int myN = -1;
    float s1x = 0.0f, s1y = 0.0f;
    if (lane < 16) {
        int n = group * 16 + lane;
        int b = -1;
        if (n < N) {
            b = bidx[n];
            float s0x = scale0[2 * b] * FSf;
            float s0y = scale0[2 * b + 1] * FSf;
            pX[wave][lane] = (int)rintf(fc0[2 * n] / s0x - 0.5f);
            pY[wave][lane] = (int)rintf(fc0[2 * n + 1] / s0y - 0.5f);
            myN = n;
            s1x = scale1[2 * b] * FSf;
            s1y = scale1[2 * b + 1] * FSf;
        }
        pB[wave][lane] = b;
    }
    __syncthreads();

    // Probe phase: 49 taps x 16 points = 784 (tap,pt) pairs spread over 32 lanes.
    for (int j = lane; j < 784; j += 32) {
        int pt = j & 15, tap = j >> 4;
        int b = pB[wave][pt];
        if (b < 0) continue;
        int x = pX[wave][pt] + (tap % 7) - 3;
        int y = pY[wave][pt] + (tap / 7) - 3;
        if (x < 0 || x >= W || y < 0 || y >= H) continue;
        int pid = (b * H + y) * W + x;
        unsigned slot = ((unsigned)pid * 2654435761u) & hmask;
        for (;;) {
            int k = keys[slot];
            if (k == pid) {
                float2 v = vals[slot];
                Vw[wave][tap * 32 + pt]      = v.x;   // channel 0 row
                Vw[wave][tap * 32 + 16 + pt] = v.y;   // channel 1 row
                occ[wave][tap] = 1;
                break;
            }
            if (k == HASH_EMPTY) break;
            slot = (slot + 1) & hmask;
        }
    }
    __syncthreads();

    // WMMA accumulation: D(16x16) rows = channels (0..7 used), cols = points.
    int half = lane >> 4;      // 0: K0/K1 (tap t), 1: K2/K3 (tap t+1)
    int r    = lane & 15;      // A: row M; B: column (point)
    v8f acc = {0.f, 0.f, 0.f, 0.f, 0.f, 0.f, 0.f, 0.f};
    for (int t = 0; t < 50; t += 2) {
        int any = occ[wave][t] | occ[wave][t + 1];
        if (__builtin_amdgcn_readfirstlane(any)) {   // scalar branch: EXEC stays all-ones
            int tt = t + half;
            v2f a, bf;
            a.x  = Tlds[tt * 32 + r * 2];            // A VGPR0: K=0 / K=2
            a.y  = Tlds[tt * 32 + r * 2 + 1];        // A VGPR1: K=1 / K=3
            bf.x = Vw[wave][tt * 32 + r];            // B VGPR0: rows K=0 / K=2 (ch0)
            bf.y = Vw[wave][tt * 32 + 16 + r];       // B VGPR1: rows K=1 / K=3 (ch1)
            acc = __builtin_amdgcn_wmma_f32_16x16x4_f32(
                      false, a, false, bf, (short)0, acc, false, false);
        }
    }

    // Epilogue: lanes 0-15 hold D rows 0..7 (channels) for point 'lane'.
    if (lane < 16 && myN >= 0) {
        float o0 = b4[0], o1 = b4[1];
        for (int o = 0; o < 8; ++o) {
            float h = acc[o] + c3g[o];
            float g = gelu_tanh(h);
            o0 += w4[o] * g;          // w4[0][o]
            o1 += w4[8 + o] * g;      // w4[1][o]
        }
        float r0 = Vw[wave][24 * 32 + lane];        // self tap (dy=0,dx=0) = residual grid
        float r1 = Vw[wave][24 * 32 + 16 + lane];
        out[2 * myN]     = (o0 + r0) * s1x;
        out[2 * myN + 1] = (o1 + r1) * s1y;
    }
}

// ---------------------------------------------------------------------------
extern "C" void kernel_launch(void* const* d_in, const int* in_sizes, int n_in,
                              void* d_out, int out_size, void* d_ws, size_t ws_size,
                              hipStream_t stream) {
    const float* fc0    = (const float*)d_in[0];
    const float* fc1    = (const float*)d_in[1];
    const int*   bidx   = (const int*)d_in[2];
    const float* scale0 = (const float*)d_in[3];
    const float* scale1 = (const float*)d_in[4];
    const float* w1     = (const float*)d_in[5];
    const float* b1     = (const float*)d_in[6];
    const float* w2     = (const float*)d_in[7];
    const float* b2     = (const float*)d_in[8];
    const float* w3     = (const float*)d_in[9];
    const float* b3     = (const float*)d_in[10];
    const float* w4     = (const float*)d_in[11];
    const float* b4     = (const float*)d_in[12];
    // d_in[13] = batch_size (unused)
    const int*   h0p    = (const int*)d_in[14];
    const int*   w0p    = (const int*)d_in[15];
    const int*   fsp    = (const int*)d_in[16];

    int N = in_sizes[0] / 2;

    // Hash capacity: prefer 2^19 (load factor 0.15), shrink if workspace is tight.
    size_t cap = (size_t)1 << 19;
    while (cap > ((size_t)1 << 17) && (8192 + 12 * cap) > ws_size) cap >>= 1;
    unsigned hmask = (unsigned)(cap - 1);

    float*  Tg   = (float*)d_ws;                                   // 1600 floats
    float*  c3g  = Tg + 1600;                                      // 8 floats
    int*    keys = (int*)((char*)d_ws + 8192);                     // cap ints
    float2* vals = (float2*)((char*)d_ws + 8192 + 4 * cap);        // cap float2
    float*  out  = (float*)d_out;

    cpr_prep<<<1, 256, 0, stream>>>(w1, b1, w2, b2, w3, b3, Tg, c3g);
    cpr_clear<<<(int)((cap + 255) / 256), 256, 0, stream>>>(keys, (int)cap);
    cpr_insert<<<(N + 255) / 256, 256, 0, stream>>>(fc0, fc1, bidx, scale0, scale1,
                                                    keys, vals, N, h0p, w0p, fsp, hmask);
    int ngroups = (N + 15) / 16;
    int nblocks = (ngroups + WPB - 1) / WPB;
    cpr_main<<<nblocks, 256, 0, stream>>>(fc0, bidx, scale0, scale1, w4, b4,
                                          Tg, c3g, keys, vals, out, N,
                                          h0p, w0p, fsp, hmask);
}